// VisualAttention_3977139716515
// MI455X (gfx1250) — compile-verified
//
#include <hip/hip_runtime.h>
#include <math.h>

typedef float v2f __attribute__((ext_vector_type(2)));
typedef float v8f __attribute__((ext_vector_type(8)));

#define B_   64
#define T_   1024
#define C_   256
#define P_   784      // 28*28 ; 25 tiles of 32 (last half-padded)
#define HID_ 512
#define F_   (T_ + C_)           // 1280 rows of W1
#define OUTD_ (C_ + T_)

// Packed B layout: Bp[(k>>2)*(HID*4) + h*4 + (k&3)] = W1[k*HID + h]
// -> per lane, the (k, k+1) pair needed by one WMMA fragment is a contiguous b64.

// ---------------------------------------------------------------------------
// Kernel 0: pack W1 (1280 x 512) into Bp.
// ---------------------------------------------------------------------------
__global__ __launch_bounds__(256) void va_pack_kernel(const float* __restrict__ W1,
                                                      float* __restrict__ Bp) {
    int g = blockIdx.x * 256 + threadIdx.x;       // 0 .. F_*HID_-1
    int k = g >> 9;                               // g / 512
    int h = g & 511;
    Bp[(k >> 2) * (HID_ * 4) + (h << 2) + (k & 3)] = W1[g];   // coalesced read
}

// ---------------------------------------------------------------------------
// Kernel 1 (WMMA): tc[b,h] = b1[h] + text[b,:] @ W1[:T,h]
// grid = (HID/16, B/16), one wave (32 threads) per 16x16 tile, K = 1024.
// A pairs loaded directly from text (contiguous b64 per lane).
// ---------------------------------------------------------------------------
__global__ __launch_bounds__(32) void va_tc_kernel(const float* __restrict__ text,
                                                   const float* __restrict__ Bp,
                                                   const float* __restrict__ b1,
                                                   float* __restrict__ tc) {
    const int lane  = threadIdx.x;
    const int ml    = lane & 15;
    const int kh1   = lane >> 4;          // 0 or 1
    const int khalf = kh1 << 1;           // 0 or 2
    const int h     = blockIdx.x * 16 + ml;
    const int b0    = blockIdx.y * 16;

    const float* arow = text + (size_t)(b0 + ml) * T_ + khalf;   // A row for this lane
    const float* bcol = Bp + (size_t)h * 4 + khalf;              // + (k>>2)*2048

    v8f acc = {};
    for (int kb = 0; kb < T_; kb += 4) {
        v2f a  = *(const v2f*)(arow + kb);                       // k = kb+khalf, +1
        v2f bv = *(const v2f*)(bcol + (kb >> 2) * (HID_ * 4));
        acc = __builtin_amdgcn_wmma_f32_16x16x4_f32(
            false, a, false, bv, (short)0, acc, false, false);
    }
    float bias = b1[h];
#pragma unroll
    for (int r = 0; r < 8; ++r)                                  // row b = b0 + r + kh1*8
        tc[(size_t)(b0 + r + kh1 * 8) * HID_ + h] = acc[r] + bias;
}

// ---------------------------------------------------------------------------
// Kernel 2: fused pixels@W1pix (+tc, ReLU, @W2, +b2) -> scores[b,p]
// grid = (ceil(P/32), B), block = 128 threads = 4 waves.
// Each wave: M-tile of 32 p (2 A tiles) x 128 h (8 N-tiles), K = C = 256.
// Each packed-B b64 load feeds two WMMAs.
// ---------------------------------------------------------------------------
__global__ __launch_bounds__(128) void va_score_kernel(const float* __restrict__ vision,
                                                       const float* __restrict__ Bp,
                                                       const float* __restrict__ W2,
                                                       const float* __restrict__ b2,
                                                       const float* __restrict__ tc,
                                                       float* __restrict__ scores) {
    __shared__ float As[2][C_ * 16];     // As[mt][c*16 + m] = pixels[b, p0+mt*16+m, c]
    __shared__ float sbuf[4][32];

    const int tid = threadIdx.x;
    const int b   = blockIdx.y;
    const int p0  = blockIdx.x * 32;

    // Stage A tiles (vision is (B,C,P): pixels[b,p,c] = vision[b*C*P + c*P + p])
    const float* vb = vision + (size_t)b * (C_ * P_);
#pragma unroll
    for (int i = 0; i < 64; ++i) {
        int idx = tid + i * 128;         // enumerates (c, m)
        int m = idx & 31;
        int c = idx >> 5;
        int p = p0 + m;
        As[m >> 4][c * 16 + (m & 15)] = (p < P_) ? vb[c * P_ + p] : 0.f;
    }
    __syncthreads();

    const int wave  = tid >> 5;
    const int lane  = tid & 31;
    const int ml    = lane & 15;
    const int kh1   = lane >> 4;
    const int khalf = kh1 << 1;
    const int hbase = wave * 128;
    const int hcol  = hbase + ml;

    v8f acc[2][8];
    v8f zero = {};
#pragma unroll
    for (int mt = 0; mt < 2; ++mt)
#pragma unroll
        for (int nt = 0; nt < 8; ++nt) acc[mt][nt] = zero;

    // pixel rows of W1 start at k = T_ -> packed group offset T_/4 = 256
    const float* bcol = Bp + (size_t)(T_ >> 2) * (HID_ * 4) + hcol * 4 + khalf;

    for (int kb = 0; kb < C_; kb += 4) {
        v2f a0, a1;
        a0.x = As[0][(kb + khalf)     * 16 + ml];
        a0.y = As[0][(kb + khalf + 1) * 16 + ml];
        a1.x = As[1][(kb + khalf)     * 16 + ml];
        a1.y = As[1][(kb + khalf + 1) * 16 + ml];
        const float* bk = bcol + (kb >> 2) * (HID_ * 4);
#pragma unroll
        for (int nt = 0; nt < 8; ++nt) {
            v2f bv = *(const v2f*)(bk + nt * 64);    // next N-tile: +16 h = +64 floats
            acc[0][nt] = __builtin_amdgcn_wmma_f32_16x16x4_f32(
                false, a0, false, bv, (short)0, acc[0][nt], false, false);
            acc[1][nt] = __builtin_amdgcn_wmma_f32_16x16x4_f32(
                false, a1, false, bv, (short)0, acc[1][nt], false, false);
        }
    }

    // Epilogue: hidden = relu(acc + tc), s_p = sum_h hidden*W2[h]
    float s[2][8] = {};
#pragma unroll
    for (int nt = 0; nt < 8; ++nt) {
        int h = hbase + nt * 16 + ml;
        float tcv = tc[b * HID_ + h];
        float w2v = W2[h];
#pragma unroll
        for (int mt = 0; mt < 2; ++mt)
#pragma unroll
            for (int r = 0; r < 8; ++r) {
                float hid = fmaxf(acc[mt][nt][r] + tcv, 0.f);
                s[mt][r] = fmaf(hid, w2v, s[mt][r]);
            }
    }
#pragma unroll
    for (int mt = 0; mt < 2; ++mt)
#pragma unroll
        for (int r = 0; r < 8; ++r)
#pragma unroll
            for (int off = 1; off < 16; off <<= 1)
                s[mt][r] += __shfl_xor(s[mt][r], off, 32);

    if (ml == 0) {                       // lanes 0 and 16
#pragma unroll
        for (int mt = 0; mt < 2; ++mt)
#pragma unroll
            for (int r = 0; r < 8; ++r)
                sbuf[wave][mt * 16 + kh1 * 8 + r] = s[mt][r];
    }
    __syncthreads();
    if (tid < 32) {
        float v = sbuf[0][tid] + sbuf[1][tid] + sbuf[2][tid] + sbuf[3][tid] + b2[0];
        int p = p0 + tid;
        if (p < P_) scores[b * P_ + p] = v;
    }
}

// ---------------------------------------------------------------------------
// Kernel 3: softmax over P, pooled[b,c] = sum_p pixels[b,p,c]*w[p], concat text.
// ---------------------------------------------------------------------------
__global__ __launch_bounds__(256) void va_pool_kernel(const float* __restrict__ text,
                                                      const float* __restrict__ vision,
                                                      const float* __restrict__ scores,
                                                      float* __restrict__ out) {
    __shared__ float wts[P_];
    __shared__ float red[8];

    const int tid  = threadIdx.x;
    const int b    = blockIdx.x;
    const int lane = tid & 31;
    const int wave = tid >> 5;
    const float* sc = scores + b * P_;

    float mx = -INFINITY;
    for (int i = tid; i < P_; i += 256) mx = fmaxf(mx, sc[i]);
    for (int off = 16; off >= 1; off >>= 1) mx = fmaxf(mx, __shfl_xor(mx, off, 32));
    if (lane == 0) red[wave] = mx;
    __syncthreads();
    mx = red[0];
#pragma unroll
    for (int wv = 1; wv < 8; ++wv) mx = fmaxf(mx, red[wv]);
    __syncthreads();

    float sum = 0.f;
    for (int i = tid; i < P_; i += 256) {
        float e = expf(sc[i] - mx);
        wts[i] = e;
        sum += e;
    }
    for (int off = 16; off >= 1; off >>= 1) sum += __shfl_xor(sum, off, 32);
    if (lane == 0) red[wave] = sum;
    __syncthreads();
    float tot = red[0];
#pragma unroll
    for (int wv = 1; wv < 8; ++wv) tot += red[wv];
    float inv = 1.f / tot;
    for (int i = tid; i < P_; i += 256) wts[i] *= inv;
    __syncthreads();

    const float* vb = vision + (size_t)b * (C_ * P_);
    for (int c = wave; c < C_; c += 8) {
        float acc = 0.f;
        const float* row = vb + c * P_;
        for (int p = lane; p < P_; p += 32)
            acc = fmaf(row[p], wts[p], acc);
        for (int off = 16; off >= 1; off >>= 1) acc += __shfl_xor(acc, off, 32);
        if (lane == 0) out[b * OUTD_ + c] = acc;
    }

    const float* tr = text + b * T_;
    float* ob = out + b * OUTD_ + C_;
    for (int t = tid; t < T_; t += 256) ob[t] = tr[t];
}

// ---------------------------------------------------------------------------
extern "C" void kernel_launch(void* const* d_in, const int* in_sizes, int n_in,
                              void* d_out, int out_size, void* d_ws, size_t ws_size,
                              hipStream_t stream) {
    const float* text   = (const float*)d_in[0];
    const float* vision = (const float*)d_in[1];
    const float* W1     = (const float*)d_in[2];
    const float* b1     = (const float*)d_in[3];
    const float* W2     = (const float*)d_in[4];
    const float* b2     = (const float*)d_in[5];
    float* out = (float*)d_out;

    float* Bp     = (float*)d_ws;            // F_*HID floats (2.56 MB), packed W1
    float* tc     = Bp + F_ * HID_;          // B*HID floats
    float* scores = tc + B_ * HID_;          // B*P floats

    va_pack_kernel<<<dim3((F_ * HID_) / 256), dim3(256), 0, stream>>>(W1, Bp);
    va_tc_kernel<<<dim3(HID_ / 16, B_ / 16), dim3(32), 0, stream>>>(text, Bp, b1, tc);
    va_score_kernel<<<dim3((P_ + 31) / 32, B_), dim3(128), 0, stream>>>(vision, Bp, W2, b2, tc, scores);
    va_pool_kernel<<<dim3(B_), dim3(256), 0, stream>>>(text, vision, scores, out);
}